// S4Lite_57226144252201
// MI455X (gfx1250) — compile-verified
//
#include <hip/hip_runtime.h>
#include <hip/hip_bf16.h>

typedef __attribute__((ext_vector_type(2))) float v2f;
typedef __attribute__((ext_vector_type(8))) float v8f;

#define DMODEL 1024
#define DSTATE 64
#define SEQ    4096
#define NROWS  32768          // 8 * 4096 flattened (b, s) rows
#define KCH    128            // K-chunk for GEMM1
#define W1PAD  65             // padded LDS row stride for W_in tile
#define CHLEN  128            // scan chunk length
#define NCHUNK (SEQ / CHLEN)  // 32 chunks per batch
#define LN_EPS 1e-5f

// ---------------------------------------------------------------------------
// K1: B_seq = x @ W_in[:64].T + b_in[:64]   (only the B half of BC is used)
// grid: 256 blocks x 256 thr; each block: 128 rows, each wave a 16-row strip.
// ---------------------------------------------------------------------------
__global__ __launch_bounds__(256) void k1_gemm_in(
    const float* __restrict__ x,      // (32768, 1024)
    const float* __restrict__ W_in,   // (128, 1024)
    const float* __restrict__ b_in,   // (128)
    float* __restrict__ Bseq)         // (32768, 64)
{
  extern __shared__ float ldsw[];     // KCH * W1PAD floats (~33 KB)
  const int t    = threadIdx.x;
  const int lane = t & 31;
  const int wave = t >> 5;
  const int rowBase = blockIdx.x * 128 + wave * 16;

  const int m   = lane & 15;          // A-matrix row (M)
  const int kb  = (lane >> 4) << 1;   // K sub-offset: lanes 0-15 -> 0, 16-31 -> 2
  const int nlo = lane & 15;          // B/C column within tile

  v8f acc[4];
  #pragma unroll
  for (int nt = 0; nt < 4; ++nt) {
    float bias = b_in[nt * 16 + nlo];
    acc[nt] = (v8f){bias, bias, bias, bias, bias, bias, bias, bias};
  }

  for (int kc = 0; kc < DMODEL; kc += KCH) {
    __syncthreads();
    // stage transposed W chunk: ldsw[k*W1PAD + n] = W_in[n][kc+k]
    #pragma unroll
    for (int i = 0; i < (KCH * DSTATE) / 256; ++i) {
      int flat = i * 256 + t;
      int k = flat & (KCH - 1);
      int n = flat >> 7;
      ldsw[k * W1PAD + n] = W_in[n * DMODEL + kc + k];
    }
    __syncthreads();

    const float* xrow = x + (size_t)(rowBase + m) * DMODEL + kc;
    for (int k0 = 0; k0 < KCH; k0 += 4) {
      v2f a;
      a.x = xrow[k0 + kb];
      a.y = xrow[k0 + kb + 1];
      #pragma unroll
      for (int nt = 0; nt < 4; ++nt) {
        int n = nt * 16 + nlo;
        v2f b;
        b.x = ldsw[(k0 + kb)     * W1PAD + n];
        b.y = ldsw[(k0 + kb + 1) * W1PAD + n];
        acc[nt] = __builtin_amdgcn_wmma_f32_16x16x4_f32(
            false, a, false, b, (short)0, acc[nt], false, false);
      }
    }
  }

  const int rowOff = (lane >> 4) * 8;   // C/D layout: lanes 16-31 hold M+8
  #pragma unroll
  for (int nt = 0; nt < 4; ++nt) {
    int col = nt * 16 + nlo;
    #pragma unroll
    for (int r = 0; r < 8; ++r) {
      Bseq[(size_t)(rowBase + r + rowOff) * DSTATE + col] = acc[nt][r];
    }
  }
}

// ---------------------------------------------------------------------------
// K2a: chunk-local scans (zero-init), in place. grid 256 x 64 threads.
// ---------------------------------------------------------------------------
__global__ __launch_bounds__(64) void k2_scan_local(
    const float* __restrict__ A_log,
    float* __restrict__ Bseq,            // in: B_seq; out: local states
    float* __restrict__ carryLocal)      // (256, 64)
{
  const int n    = threadIdx.x;
  const int cidx = blockIdx.x;           // b*32 + chunk (rows tile contiguously)
  const float A  = -__expf(A_log[n]);
  float s = 0.f;
  size_t base = (size_t)cidx * CHLEN * DSTATE + n;
  for (int tt = 0; tt < CHLEN; ++tt) {
    size_t idx = base + (size_t)tt * DSTATE;
    s = A * s + Bseq[idx];
    Bseq[idx] = s;
  }
  carryLocal[cidx * DSTATE + n] = s;
}

// ---------------------------------------------------------------------------
// K2b: combine carries across 32 chunks; also emits final_state.
// A^128 = exp(128*A_log) (even power -> positive).  1 block x 512 threads.
// ---------------------------------------------------------------------------
__global__ __launch_bounds__(512) void k2_scan_carry(
    const float* __restrict__ A_log,
    const float* __restrict__ state0,     // (8, 64)
    const float* __restrict__ carryLocal, // (256, 64)
    float* __restrict__ carryIn,          // (256, 64)
    float* __restrict__ final_state)      // (8, 64)
{
  const int n = threadIdx.x & 63;
  const int b = threadIdx.x >> 6;
  const float Ach = __expf((float)CHLEN * A_log[n]);
  float c = state0[b * DSTATE + n];
  for (int j = 0; j < NCHUNK; ++j) {
    int idx = (b * NCHUNK + j) * DSTATE + n;
    carryIn[idx] = c;
    c = Ach * c + carryLocal[idx];
  }
  final_state[b * DSTATE + n] = c;
}

// ---------------------------------------------------------------------------
// K3: out = LN( states@W_out.T + b_out + D*x ); carry fix fused into A-stage.
// grid 1024 x 256 thr; each block: 32 rows x 1024 cols. LDS ~139 KB.
// ---------------------------------------------------------------------------
__global__ __launch_bounds__(256) void k3_gemm_out_ln(
    const float* __restrict__ states,   // (32768, 64) local states
    const float* __restrict__ carryIn,  // (256, 64)
    const float* __restrict__ A_log,    // (64)
    const float* __restrict__ x,        // (32768, 1024)
    const float* __restrict__ W_out,    // (1024, 64)
    const float* __restrict__ b_out,    // (1024)
    const float* __restrict__ Dv,       // (1024)
    const float* __restrict__ gamma,
    const float* __restrict__ beta,
    float* __restrict__ out)            // (32768, 1024)
{
  extern __shared__ float lds[];
  float* a_tile = lds;                   // 32 * 68
  float* out_s  = a_tile + 32 * 68;      // 32 * 1025
  float* psum   = out_s + 32 * 1025;     // 32 * 8
  float* psq    = psum + 256;            // 32 * 8
  float* mu_s   = psq + 256;             // 32
  float* rs_s   = mu_s + 32;             // 32

  const int t    = threadIdx.x;
  const int lane = t & 31;
  const int wave = t >> 5;
  const int rowBase = blockIdx.x * 32;

  // 1) stage states tile with scan carry fix: s += (-1)^p * exp(p*A_log) * carry
  #pragma unroll
  for (int i = 0; i < 8; ++i) {
    int flat = i * 256 + t;              // 0..2047
    int r = flat >> 6;                   // 0..31
    int n = flat & 63;
    int grow = rowBase + r;
    int p = (grow & (CHLEN - 1)) + 1;
    float f = __expf((float)p * A_log[n]);
    if (p & 1) f = -f;
    float val = states[(size_t)grow * DSTATE + n] +
                f * carryIn[(grow >> 7) * DSTATE + n];
    a_tile[r * 68 + n] = val;
  }
  __syncthreads();

  // 2) WMMA GEMM2: waves 0-3 -> row strip 0, waves 4-7 -> strip 1
  const int strip = wave >> 2;
  const int wq    = wave & 3;
  const int mrow  = strip * 16 + (lane & 15);
  const int kb    = (lane >> 4) << 1;
  const int nlo   = lane & 15;
  const int rowOff = (lane >> 4) * 8;
  for (int ct = 0; ct < 16; ++ct) {
    int col0 = (ct * 4 + wq) * 16;
    v8f acc = (v8f){0.f, 0.f, 0.f, 0.f, 0.f, 0.f, 0.f, 0.f};
    #pragma unroll
    for (int k0 = 0; k0 < DSTATE; k0 += 4) {
      v2f a, b;
      a.x = a_tile[mrow * 68 + k0 + kb];
      a.y = a_tile[mrow * 68 + k0 + kb + 1];
      int d = col0 + nlo;
      b.x = W_out[d * DSTATE + k0 + kb];
      b.y = W_out[d * DSTATE + k0 + kb + 1];
      acc = __builtin_amdgcn_wmma_f32_16x16x4_f32(
          false, a, false, b, (short)0, acc, false, false);
    }
    #pragma unroll
    for (int r = 0; r < 8; ++r) {
      out_s[(strip * 16 + r + rowOff) * 1025 + col0 + nlo] = acc[r];
    }
  }
  __syncthreads();

  // 3) epilogue: + b_out + D*x  (coalesced x read, once)
  #pragma unroll 4
  for (int i = 0; i < 128; ++i) {
    int flat = i * 256 + t;
    int r = flat >> 10;
    int c = flat & 1023;
    out_s[r * 1025 + c] += b_out[c] + Dv[c] * x[(size_t)(rowBase + r) * DMODEL + c];
  }
  __syncthreads();

  // 4) LayerNorm stats: 8 partials per row, then per-row combine
  {
    int r = t >> 3;
    int g = t & 7;
    float s = 0.f, sq = 0.f;
    for (int i = 0; i < 128; ++i) {
      float v = out_s[r * 1025 + g + 8 * i];
      s += v;
      sq += v * v;
    }
    psum[r * 8 + g] = s;
    psq[r * 8 + g] = sq;
  }
  __syncthreads();
  if (t < 32) {
    float s = 0.f, sq = 0.f;
    #pragma unroll
    for (int g = 0; g < 8; ++g) { s += psum[t * 8 + g]; sq += psq[t * 8 + g]; }
    float mu = s * (1.f / 1024.f);
    float var = sq * (1.f / 1024.f) - mu * mu;
    mu_s[t] = mu;
    rs_s[t] = rsqrtf(var + LN_EPS);
  }
  __syncthreads();

  // 5) normalize + single coalesced store to HBM
  #pragma unroll 4
  for (int i = 0; i < 128; ++i) {
    int flat = i * 256 + t;
    int r = flat >> 10;
    int c = flat & 1023;
    float v = (out_s[r * 1025 + c] - mu_s[r]) * rs_s[r] * gamma[c] + beta[c];
    out[(size_t)(rowBase + r) * DMODEL + c] = v;
  }
}

// ---------------------------------------------------------------------------
extern "C" void kernel_launch(void* const* d_in, const int* in_sizes, int n_in,
                              void* d_out, int out_size, void* d_ws, size_t ws_size,
                              hipStream_t stream) {
  const float* x      = (const float*)d_in[0];
  const float* state0 = (const float*)d_in[1];
  const float* A_log  = (const float*)d_in[2];
  const float* Dv     = (const float*)d_in[3];
  const float* W_in   = (const float*)d_in[4];
  const float* b_in   = (const float*)d_in[5];
  const float* W_out  = (const float*)d_in[6];
  const float* b_out  = (const float*)d_in[7];
  const float* gamma  = (const float*)d_in[8];
  const float* beta   = (const float*)d_in[9];

  float* out = (float*)d_out;
  float* final_state = out + (size_t)NROWS * DMODEL;   // tuple output #2

  float* Bseq       = (float*)d_ws;                    // 32768*64
  float* carryLocal = Bseq + (size_t)NROWS * DSTATE;   // 256*64
  float* carryIn    = carryLocal + 8 * NCHUNK * DSTATE;

  size_t lds1 = (size_t)KCH * W1PAD * sizeof(float);
  k1_gemm_in<<<NROWS / 128, 256, lds1, stream>>>(x, W_in, b_in, Bseq);

  k2_scan_local<<<8 * NCHUNK, 64, 0, stream>>>(A_log, Bseq, carryLocal);
  k2_scan_carry<<<1, 512, 0, stream>>>(A_log, state0, carryLocal, carryIn, final_state);

  size_t lds3 = (size_t)(32 * 68 + 32 * 1025 + 256 + 256 + 32 + 32) * sizeof(float);
  k3_gemm_out_ln<<<NROWS / 32, 256, lds3, stream>>>(
      Bseq, carryIn, A_log, x, W_out, b_out, Dv, gamma, beta, out);
}